// StructuredPruningIF_31636729102640
// MI455X (gfx1250) — compile-verified
//
#include <hip/hip_runtime.h>
#include <stdint.h>

// Problem geometry (fixed by the reference).
#define TT 4
#define BB 32
#define CC 256
#define HW 1024                       // 32*32
#define IMGS (TT*BB)                  // 128 images of shape (CC, 32, 32)
#define TSTRIDE4 ((BB*CC*HW)/4)       // float4 stride between time steps = 2,097,152
#define ELEMS_PER_CH ((float)(TT*BB*HW))  // 131072 elements reduced per channel

typedef __attribute__((ext_vector_type(2))) float v2f;
typedef __attribute__((ext_vector_type(8))) float v8f;

// ---------------------------------------------------------------------------
// CDNA5 async global->LDS copy (ASYNCcnt-tracked). GVS addressing:
//   mem_addr = SGPR64 base + signed 32-bit VGPR offset. VDST = LDS byte addr.
// ---------------------------------------------------------------------------
__device__ __forceinline__ void async_copy_b128(unsigned lds_addr, unsigned gbyte_off,
                                                const void* base) {
  asm volatile("global_load_async_to_lds_b128 %0, %1, %2"
               :: "v"(lds_addr), "v"(gbyte_off), "s"(base)
               : "memory");
}
__device__ __forceinline__ void wait_asynccnt_le1() {
  asm volatile("s_wait_asynccnt 0x1" ::: "memory");
}
__device__ __forceinline__ void wait_asynccnt_0() {
  asm volatile("s_wait_asynccnt 0x0" ::: "memory");
}
__device__ __forceinline__ void wait_dscnt_0() {
  asm volatile("s_wait_dscnt 0x0" ::: "memory");
}

// ---------------------------------------------------------------------------
// Kernel 1: per-channel sum(x^2) and count(x>0).
// One block per channel. Each image contributes a contiguous 4 KB chunk
// (1024 floats) at element offset (n*CC + c)*HW. 256 threads * float4 = one
// chunk per iteration, streamed through LDS with depth-2 async double
// buffering. Each thread reads back only the LDS slot it filled, so per-wave
// ASYNCcnt waits suffice (no block barrier in the stream loop).
//
// Final 256->1 block reduction uses the matrix pipe: with A == ones(16x4),
// V_WMMA_F32_16X16X4_F32 computes D[m,n] = sum_k B[k,n] + C[m,n]; each B
// operand (2 VGPRs) carries 64 partials, so 4 chained WMMAs fold all 256
// partials into 16 column sums (row 0 of D = VGPR0, lanes 0..15), finished
// with 4 shfl_xor steps. Executed by wave 0 only (EXEC all-ones, as WMMA
// requires).
// ---------------------------------------------------------------------------
__global__ __launch_bounds__(256) void reduce_kernel(const float* __restrict__ x,
                                                     float* __restrict__ sumsq,
                                                     float* __restrict__ cnt) {
  __shared__ float4 stage[2][256];   // 2 x 4 KB staging buffers
  __shared__ float  redA[256];       // sum-of-squares partials
  __shared__ float  redB[256];       // positive-count partials

  const unsigned tid = threadIdx.x;
  const unsigned c   = blockIdx.x;

  const unsigned lds_slot0 = (unsigned)(uintptr_t)&stage[0][tid];
  const unsigned lds_slot1 = (unsigned)(uintptr_t)&stage[1][tid];

  float acc = 0.0f;   // sum of squares
  float pos = 0.0f;   // count of positives (exact in f32: <= 131072 < 2^24)

  // Prologue: image 0 -> buffer 0.
  {
    unsigned g0 = ((0u * CC + c) * HW + tid * 4u) * 4u;   // byte offset
    async_copy_b128(lds_slot0, g0, x);
  }

  #pragma unroll 1
  for (int n = 0; n < IMGS; ++n) {
    if (n + 1 < IMGS) {
      // Our previous DS read of the target buffer must have returned before
      // the async engine may overwrite it; then issue the next tile.
      wait_dscnt_0();
      unsigned g = (((unsigned)(n + 1) * CC + c) * HW + tid * 4u) * 4u;
      async_copy_b128(((n + 1) & 1) ? lds_slot1 : lds_slot0, g, x);
      wait_asynccnt_le1();   // loads complete in order -> tile n has landed
    } else {
      wait_asynccnt_0();
    }
    float4 v = stage[n & 1][tid];
    acc += v.x * v.x + v.y * v.y + v.z * v.z + v.w * v.w;
    pos += (v.x > 0.0f ? 1.0f : 0.0f) + (v.y > 0.0f ? 1.0f : 0.0f) +
           (v.z > 0.0f ? 1.0f : 0.0f) + (v.w > 0.0f ? 1.0f : 0.0f);
  }

  redA[tid] = acc;
  redB[tid] = pos;
  __syncthreads();

  if (tid < 32) {               // wave 0, all 32 lanes active -> EXEC all ones
    v2f ones;
    ones.x = 1.0f;
    ones.y = 1.0f;
    v8f dacc = {};
    v8f dpos = {};
    #pragma unroll
    for (int i = 0; i < 4; ++i) {
      v2f ba, bb;
      ba.x = redA[i * 64 + tid];
      ba.y = redA[i * 64 + 32 + tid];
      bb.x = redB[i * 64 + tid];
      bb.y = redB[i * 64 + 32 + tid];
      dacc = __builtin_amdgcn_wmma_f32_16x16x4_f32(false, ones, false, ba,
                                                   (short)0, dacc, false, false);
      dpos = __builtin_amdgcn_wmma_f32_16x16x4_f32(false, ones, false, bb,
                                                   (short)0, dpos, false, false);
    }
    // Row 0 of D: VGPR0 lanes 0..15 hold the 16 column sums (lanes 16..31
    // hold row M=8, which is identical since every row of ones*B matches).
    float va = dacc[0];
    float vp = dpos[0];
    va += __shfl_xor(va, 1);  vp += __shfl_xor(vp, 1);
    va += __shfl_xor(va, 2);  vp += __shfl_xor(vp, 2);
    va += __shfl_xor(va, 4);  vp += __shfl_xor(vp, 4);
    va += __shfl_xor(va, 8);  vp += __shfl_xor(vp, 8);
    if (tid == 0) {
      sumsq[c] = va;
      cnt[c]   = vp;
    }
  }
}

// ---------------------------------------------------------------------------
// Kernel 2: importance + exact top_k(-importance, 64) replica via ranking.
// Channel c is pruned iff fewer than 64 channels are strictly "more prunable"
// (smaller importance, ties broken toward the lower index) -- identical
// selection set to jax.lax.top_k on negated importance.
// ---------------------------------------------------------------------------
__global__ __launch_bounds__(256) void mask_kernel(const float* __restrict__ sumsq,
                                                   const float* __restrict__ cnt,
                                                   float* __restrict__ mask) {
  __shared__ float imp[CC];
  const int c = threadIdx.x;
  const float im = 0.7f * sqrtf(sumsq[c]) + 0.3f * (cnt[c] * (1.0f / ELEMS_PER_CH));
  imp[c] = im;
  __syncthreads();

  int rank = 0;
  #pragma unroll 8
  for (int j = 0; j < CC; ++j) {
    const float o = imp[j];
    rank += (o < im) || (o == im && j < c);
  }
  mask[c] = (rank < (CC / 4)) ? 0.0f : 1.0f;   // 64 pruned channels
}

// ---------------------------------------------------------------------------
// Kernel 3: integrate-and-fire scan over T=4, one float4 spatial vector per
// thread. Same flat layout for input and output: element(t,b,c,h,w) lives at
// ((t*BB+b)*CC + c)*HW + h*32 + w. Time stride = 32 MB; the 4 loads per
// thread are independent b128s the compiler issues back-to-back.
// ---------------------------------------------------------------------------
__global__ __launch_bounds__(256) void scan_kernel(const float* __restrict__ x,
                                                   const float* __restrict__ thresh,
                                                   const float* __restrict__ maskbuf,
                                                   float* __restrict__ out) {
  const unsigned flat4 = blockIdx.x * 256u + threadIdx.x;   // 0 .. 2,097,151
  const unsigned c = (flat4 >> 8) & 255u;                   // HW/4 = 256 float4 per plane

  const float thre = thresh[0];
  const float tm   = thre * maskbuf[c];   // masked spike magnitude

  const float4* __restrict__ x4 = (const float4*)x;
  float4* __restrict__ o4       = (float4*)out;

  float4 m;
  m.x = m.y = m.z = m.w = 0.5f * thre;

  #pragma unroll
  for (int t = 0; t < TT; ++t) {
    const size_t idx = (size_t)flat4 + (size_t)t * TSTRIDE4;
    const float4 xv = x4[idx];
    m.x += xv.x; m.y += xv.y; m.z += xv.z; m.w += xv.w;
    float4 s;
    s.x = (m.x >= thre) ? tm : 0.0f;
    s.y = (m.y >= thre) ? tm : 0.0f;
    s.z = (m.z >= thre) ? tm : 0.0f;
    s.w = (m.w >= thre) ? tm : 0.0f;
    m.x -= s.x; m.y -= s.y; m.z -= s.z; m.w -= s.w;
    o4[idx] = s;
  }
}

// ---------------------------------------------------------------------------
// Launch wrapper. d_in[0] = x (f32, 33,554,432 elems), d_in[1] = thresh (f32, 1).
// d_out = spikes (f32, 33,554,432). d_ws: 3*256 floats of scratch
// (sumsq | cnt | mask), fully written before read -> no init needed.
// ---------------------------------------------------------------------------
extern "C" void kernel_launch(void* const* d_in, const int* in_sizes, int n_in,
                              void* d_out, int out_size, void* d_ws, size_t ws_size,
                              hipStream_t stream) {
  const float* x      = (const float*)d_in[0];
  const float* thresh = (const float*)d_in[1];
  float* out          = (float*)d_out;

  float* sumsq = (float*)d_ws;
  float* cnt   = sumsq + CC;
  float* mask  = cnt + CC;

  reduce_kernel<<<dim3(CC), dim3(256), 0, stream>>>(x, sumsq, cnt);
  mask_kernel<<<dim3(1), dim3(256), 0, stream>>>(sumsq, cnt, mask);
  scan_kernel<<<dim3((BB * CC * HW / 4) / 256), dim3(256), 0, stream>>>(x, thresh, mask, out);
}